// FillDiagonalsFromArray_25417616458409
// MI455X (gfx1250) — compile-verified
//
#include <hip/hip_runtime.h>
#include <cstdint>

// output[n, i, j] = input[n, |i-j|]  — batched symmetric Toeplitz expansion.
// N=64, M=2048. Pure data movement: 0.5 MB read, 1.07 GB written.
// Strategy (MI455X / gfx1250):
//   * stage the 8 KB input row in LDS via CDNA5 async global->LDS b128 copies
//     (ASYNCcnt + s_wait_asynccnt),
//   * gather from LDS (|i-j| index arithmetic in VALU),
//   * stream results out with non-temporal 128-bit stores (write-once data,
//     don't pollute the 192 MB L2).

typedef float f32x4 __attribute__((ext_vector_type(4)));

#define M_DIM 2048
#define N_BATCH 64
#define ROWS_PER_BLOCK 16
#define THREADS 256

__global__ __launch_bounds__(THREADS)
void toeplitz_expand_kernel(const float* __restrict__ src,
                            float* __restrict__ out) {
  __shared__ float srow[M_DIM];  // 8 KB: one input row, reused 16*2048 times

  const int t = threadIdx.x;                       // 0..255
  const int groups_per_batch = M_DIM / ROWS_PER_BLOCK;   // 128
  const int n  = blockIdx.x / groups_per_batch;          // 0..63
  const int i0 = (blockIdx.x % groups_per_batch) * ROWS_PER_BLOCK;

  // ---- Phase 1: async-copy input row n into LDS (2 x 16B per thread) ----
  {
    const float* g = src + (size_t)n * M_DIM + (size_t)t * 4;
    unsigned lds = (unsigned)(uintptr_t)(&srow[t * 4]);  // LDS byte offset
    asm volatile("global_load_async_to_lds_b128 %0, %1, off"
                 :: "v"(lds), "v"(g) : "memory");
    asm volatile("global_load_async_to_lds_b128 %0, %1, off"
                 :: "v"(lds + (unsigned)(THREADS * 16)),
                    "v"(g + THREADS * 4) : "memory");
    asm volatile("s_wait_asynccnt 0" ::: "memory");
  }
  __syncthreads();  // all 8 waves' async writes to LDS are now visible

  // ---- Phase 2: emit 16 output rows, NT b128 stores ----
  for (int r = 0; r < ROWS_PER_BLOCK; ++r) {
    const int i = i0 + r;
    float* orow = out + ((size_t)n * M_DIM + (size_t)i) * (size_t)M_DIM;
#pragma unroll
    for (int p = 0; p < M_DIM / (THREADS * 4); ++p) {  // 2 passes
      const int j = (p * THREADS + t) * 4;
      const int b = i - j;                 // |i-(j+k)| = |b-k|
      int d0 = b;     d0 = d0 < 0 ? -d0 : d0;
      int d1 = b - 1; d1 = d1 < 0 ? -d1 : d1;
      int d2 = b - 2; d2 = d2 < 0 ? -d2 : d2;
      int d3 = b - 3; d3 = d3 < 0 ? -d3 : d3;
      f32x4 v;
      v.x = srow[d0];
      v.y = srow[d1];
      v.z = srow[d2];
      v.w = srow[d3];
      __builtin_nontemporal_store(v, (f32x4*)(orow + j));
    }
  }
}

extern "C" void kernel_launch(void* const* d_in, const int* in_sizes, int n_in,
                              void* d_out, int out_size, void* d_ws, size_t ws_size,
                              hipStream_t stream) {
  (void)in_sizes; (void)n_in; (void)out_size; (void)d_ws; (void)ws_size;
  const float* src = (const float*)d_in[0];
  float* out = (float*)d_out;

  dim3 grid(N_BATCH * (M_DIM / ROWS_PER_BLOCK));  // 8192 blocks
  dim3 block(THREADS);                            // 8 waves (wave32)
  hipLaunchKernelGGL(toeplitz_expand_kernel, grid, block, 0, stream, src, out);
}